// NodeOperation_17815524344039
// MI455X (gfx1250) — compile-verified
//
#include <hip/hip_runtime.h>
#include <math.h>

typedef __attribute__((ext_vector_type(2))) float v2f;
typedef __attribute__((ext_vector_type(8))) float v8f;
typedef __attribute__((ext_vector_type(4))) unsigned int u32x4;
typedef __attribute__((ext_vector_type(8))) int i32x8;
typedef __attribute__((ext_vector_type(4))) int i32x4;

static constexpr int kC = 96;          // channels
static constexpr int kTM = 16;         // edge/node rows per wave tile
static constexpr int kWaves = 4;       // waves per block (128 threads)

__device__ __forceinline__ void wave_lds_fence() {
  // intra-wave LDS producer->consumer: wave32 lockstep + dscnt drain
  __asm__ volatile("s_wait_dscnt 0" ::: "memory");
}

__device__ __forceinline__ float silu_f(float x) {
  return x * __builtin_amdgcn_rcpf(1.0f + __expf(-x));
}
__device__ __forceinline__ float sigmoid_f(float x) {
  return __builtin_amdgcn_rcpf(1.0f + __expf(-x));
}

// -------- Tensor Data Mover: 1-D contiguous f32 copy, global -> LDS ----------
// Wave-uniform source/dest; one DMA per wave, tracked with TENSORcnt.
__device__ __forceinline__ void tdm_load_tile_f32(const float* gsrc,
                                                  unsigned int lds_byte_off,
                                                  int nElems) {
  const unsigned long long ga = (unsigned long long)(uintptr_t)gsrc;
  const unsigned int ga_lo = (unsigned int)(ga & 0xffffffffull);
  const unsigned int ga_hi25 = (unsigned int)((ga >> 32) & 0x1ffffffull);

  // D# group 0 (128b): count=1 | lds_addr | global_addr[56:0] | type=2
  u32x4 g0;
  g0.x = 1u;                                   // count=1, is_restore=0, gather off
  g0.y = lds_byte_off;                         // lds_addr (bytes)
  g0.z = ga_lo;                                // global_addr[31:0]  (bits 95:64)
  g0.w = ga_hi25 | (2u << 30);                 // global_addr[56:32] | type=2

  // D# group 1 (256b): data_size=4B, tensor_dim0=tile_dim0=nElems, dim1=1
  const unsigned int n = (unsigned int)nElems;
  i32x8 g1;
  g1[0] = (int)(2u << 16);                     // workgroup_mask=0, data_size=2 (4B)
  g1[1] = (int)((n & 0xffffu) << 16);          // tensor_dim0[15:0] @ bits 63:48
  g1[2] = (int)(((n >> 16) & 0xffffu) | (1u << 16)); // tensor_dim0[31:16] | tensor_dim1=1
  g1[3] = (int)((n & 0xffffu) << 16);          // tensor_dim1[31:16]=0 | tile_dim0 @ bits 127:112
  g1[4] = 1;                                   // tile_dim1=1, tile_dim2=0
  g1[5] = (int)n;                              // tensor_dim0_stride[31:0]
  g1[6] = 0;                                   // stride high bits
  g1[7] = 0;
  i32x4 gz4 = {0, 0, 0, 0};
  i32x8 gz8 = {0, 0, 0, 0, 0, 0, 0, 0};

  __builtin_amdgcn_tensor_load_to_lds(g0, g1, gz4, gz4, gz8, 0);
  __builtin_amdgcn_s_wait_tensorcnt(0);
  __asm__ volatile("" ::: "memory");           // LDS now holds the tile
}

// [16 x K](LDS, row-major stride K) @ W[K x 96](global, row-major) + bias, SiLU -> Hlds[16 x 96]
__device__ __forceinline__ void gemm_ldsA_silu(const float* __restrict__ Alds, int K,
                                               const float* __restrict__ W,
                                               const float* __restrict__ bias,
                                               float* __restrict__ Hlds, int lane) {
  const int half = lane >> 4;
  const int mrow = lane & 15;
  v8f acc[6];
#pragma unroll
  for (int nt = 0; nt < 6; ++nt) {
    float bv = bias[nt * 16 + mrow];
    acc[nt] = (v8f){bv, bv, bv, bv, bv, bv, bv, bv};
  }
  for (int k0 = 2 * half; k0 < K; k0 += 4) {
    v2f a;
    a.x = Alds[mrow * K + k0];
    a.y = Alds[mrow * K + k0 + 1];
#pragma unroll
    for (int nt = 0; nt < 6; ++nt) {
      const int n = nt * 16 + mrow;
      v2f b;
      b.x = W[(size_t)k0 * kC + n];
      b.y = W[(size_t)(k0 + 1) * kC + n];
      acc[nt] = __builtin_amdgcn_wmma_f32_16x16x4_f32(false, a, false, b,
                                                      (short)0, acc[nt], false, false);
    }
  }
#pragma unroll
  for (int nt = 0; nt < 6; ++nt) {
    const int n = nt * 16 + mrow;
#pragma unroll
    for (int v = 0; v < 8; ++v)
      Hlds[(v + 8 * half) * kC + n] = silu_f(acc[nt][v]);
  }
}

__global__ __launch_bounds__(128)
void edge_kernel(const float* __restrict__ m_ji, const float* __restrict__ coords,
                 const long long* __restrict__ eidx,
                 const float* __restrict__ Wa1, const float* __restrict__ ba1,
                 const float* __restrict__ Wa2, const float* __restrict__ ba2,
                 const float* __restrict__ Wx1, const float* __restrict__ bx1,
                 const float* __restrict__ Wx2, const float* __restrict__ bx2,
                 float* __restrict__ att_out, float* __restrict__ m_acc,
                 float* __restrict__ d_acc, int E_, int nTiles) {
  __shared__ float sM[kWaves][kTM * kC];
  __shared__ float sH[kWaves][kTM * kC];
  __shared__ float sAtt[kWaves][kTM];
  __shared__ float sPhx[kWaves][kTM];

  const int lane = threadIdx.x & 31;
  const int wib  = threadIdx.x >> 5;
  const int tile = blockIdx.x * kWaves + wib;
  if (tile >= nTiles) return;                 // wave-uniform
  const int e0 = tile * kTM;

  float* Mt = sM[wib];
  float* Ht = sH[wib];

  // DMA the contiguous 16x96 message tile into LDS via the Tensor Data Mover
  tdm_load_tile_f32(m_ji + (size_t)e0 * kC,
                    (unsigned int)(uintptr_t)Mt, kTM * kC);

  // ---------- attention MLP: silu(m@Wa1+ba1) @ Wa2 + ba2 -> sigmoid ----------
  gemm_ldsA_silu(Mt, kC, Wa1, ba1, Ht, lane);
  wave_lds_fence();
  {
    const int row = lane & 15;
    float s = ba2[0];
    for (int k = 0; k < kC; ++k) s += Ht[row * kC + k] * Wa2[k];
    const float att = sigmoid_f(s);
    if (lane < 16) {
      sAtt[wib][row] = att;
      att_out[e0 + row] = att;
    }
  }

  // ---------- phi_x MLP: silu(m@Wx1+bx1) @ Wx2 + bx2 ----------
  gemm_ldsA_silu(Mt, kC, Wx1, bx1, Ht, lane);
  wave_lds_fence();
  {
    const int row = lane & 15;
    float s = bx2[0];
    for (int k = 0; k < kC; ++k) s += Ht[row * kC + k] * Wx2[k];
    if (lane < 16) sPhx[wib][row] = s;
  }
  wave_lds_fence();

  // ---------- scatter attention-weighted messages ----------
  for (int e = 0; e < kTM; ++e) {
    const long long recv = eidx[e0 + e];
    const float a = sAtt[wib][e];
    float* dst = m_acc + (size_t)recv * kC;
    for (int c = lane; c < kC; c += 32)
      unsafeAtomicAdd(dst + c, Mt[e * kC + c] * a);
  }

  // ---------- scatter coordinate deltas (one edge per lane 0..15) ----------
  if (lane < 16) {
    const int e = lane;
    const long long recv = eidx[e0 + e];
    const long long send = eidx[(size_t)E_ + e0 + e];
    const float rx = coords[send * 3 + 0] - coords[recv * 3 + 0];
    const float ry = coords[send * 3 + 1] - coords[recv * 3 + 1];
    const float rz = coords[send * 3 + 2] - coords[recv * 3 + 2];
    const float nrm = sqrtf(rx * rx + ry * ry + rz * rz);
    const float sc  = sPhx[wib][e] / (nrm + 1.0f);
    float* dd = d_acc + (size_t)recv * 3;
    unsafeAtomicAdd(dd + 0, rx * sc);
    unsafeAtomicAdd(dd + 1, ry * sc);
    unsafeAtomicAdd(dd + 2, rz * sc);
  }
}

__global__ __launch_bounds__(128)
void node_kernel(const float* __restrict__ nf, const float* __restrict__ coords,
                 const float* __restrict__ Wn1, const float* __restrict__ bn1,
                 const float* __restrict__ Wn2, const float* __restrict__ bn2,
                 const float* __restrict__ m_acc, const float* __restrict__ d_acc,
                 float* __restrict__ nf_out, float* __restrict__ coord_out,
                 int N_, int nTiles) {
  __shared__ float sH[kWaves][kTM * kC];
  const int lane = threadIdx.x & 31;
  const int wib  = threadIdx.x >> 5;
  const int tile = blockIdx.x * kWaves + wib;
  if (tile >= nTiles) return;                 // wave-uniform
  const int n0   = tile * kTM;
  const int half = lane >> 4;
  const int mrow = lane & 15;
  float* Ht = sH[wib];

  // ---------- layer 1: concat(nf, m_acc)[16x192] @ Wn1[192x96] + bn1, SiLU ----------
  v8f acc[6];
#pragma unroll
  for (int nt = 0; nt < 6; ++nt) {
    float bv = bn1[nt * 16 + mrow];
    acc[nt] = (v8f){bv, bv, bv, bv, bv, bv, bv, bv};
  }
  const size_t rowBase = (size_t)(n0 + mrow) * kC;
  for (int k0 = 2 * half; k0 < 2 * kC; k0 += 4) {
    v2f a;   // even k0 -> the (k0,k0+1) pair never straddles the concat boundary
    if (k0 < kC) { a.x = nf[rowBase + k0];           a.y = nf[rowBase + k0 + 1]; }
    else         { a.x = m_acc[rowBase + (k0 - kC)]; a.y = m_acc[rowBase + (k0 - kC) + 1]; }
#pragma unroll
    for (int nt = 0; nt < 6; ++nt) {
      const int n = nt * 16 + mrow;
      v2f b;
      b.x = Wn1[(size_t)k0 * kC + n];
      b.y = Wn1[(size_t)(k0 + 1) * kC + n];
      acc[nt] = __builtin_amdgcn_wmma_f32_16x16x4_f32(false, a, false, b,
                                                      (short)0, acc[nt], false, false);
    }
  }
#pragma unroll
  for (int nt = 0; nt < 6; ++nt) {
    const int n = nt * 16 + mrow;
#pragma unroll
    for (int v = 0; v < 8; ++v)
      Ht[(v + 8 * half) * kC + n] = silu_f(acc[nt][v]);
  }
  wave_lds_fence();

  // ---------- layer 2: h[16x96] @ Wn2[96x96] + bn2 + residual ----------
  v8f acc2[6];
#pragma unroll
  for (int nt = 0; nt < 6; ++nt) {
    float bv = bn2[nt * 16 + mrow];
    acc2[nt] = (v8f){bv, bv, bv, bv, bv, bv, bv, bv};
  }
  for (int k0 = 2 * half; k0 < kC; k0 += 4) {
    v2f a;
    a.x = Ht[mrow * kC + k0];
    a.y = Ht[mrow * kC + k0 + 1];
#pragma unroll
    for (int nt = 0; nt < 6; ++nt) {
      const int n = nt * 16 + mrow;
      v2f b;
      b.x = Wn2[(size_t)k0 * kC + n];
      b.y = Wn2[(size_t)(k0 + 1) * kC + n];
      acc2[nt] = __builtin_amdgcn_wmma_f32_16x16x4_f32(false, a, false, b,
                                                       (short)0, acc2[nt], false, false);
    }
  }
#pragma unroll
  for (int nt = 0; nt < 6; ++nt) {
    const int n = nt * 16 + mrow;
#pragma unroll
    for (int v = 0; v < 8; ++v) {
      const size_t idx = (size_t)(n0 + v + 8 * half) * kC + n;
      nf_out[idx] = acc2[nt][v] + nf[idx];     // residual
    }
  }

  // ---------- coordinates: coords + accumulated deltas ----------
  if (lane < 16) {
    const size_t node = (size_t)(n0 + mrow);
#pragma unroll
    for (int j = 0; j < 3; ++j)
      coord_out[node * 3 + j] = coords[node * 3 + j] + d_acc[node * 3 + j];
  }
}

extern "C" void kernel_launch(void* const* d_in, const int* in_sizes, int n_in,
                              void* d_out, int out_size, void* d_ws, size_t ws_size,
                              hipStream_t stream) {
  const float*     nf   = (const float*)d_in[0];
  const float*     co   = (const float*)d_in[1];
  const float*     mji  = (const float*)d_in[2];
  const long long* eidx = (const long long*)d_in[3];   // int64 per reference
  const float* Wn1 = (const float*)d_in[4];
  const float* bn1 = (const float*)d_in[5];
  const float* Wn2 = (const float*)d_in[6];
  const float* bn2 = (const float*)d_in[7];
  const float* Wa1 = (const float*)d_in[8];
  const float* ba1 = (const float*)d_in[9];
  const float* Wa2 = (const float*)d_in[10];
  const float* ba2 = (const float*)d_in[11];
  const float* Wx1 = (const float*)d_in[12];
  const float* bx1 = (const float*)d_in[13];
  const float* Wx2 = (const float*)d_in[14];
  const float* bx2 = (const float*)d_in[15];

  const int N_ = in_sizes[0] / kC;
  const int E_ = in_sizes[2] / kC;

  float* m_acc  = (float*)d_ws;                 // [N,96]
  float* d_acc  = m_acc + (size_t)N_ * kC;      // [N,3]
  float* nf_out = (float*)d_out;                // [N,96]
  float* co_out = nf_out + (size_t)N_ * kC;     // [N,3]
  float* a_out  = co_out + (size_t)N_ * 3;      // [E,1]

  (void)hipMemsetAsync(d_ws, 0, (size_t)N_ * (kC + 3) * sizeof(float), stream);

  const int tilesE  = E_ / kTM;
  const int blocksE = (tilesE + kWaves - 1) / kWaves;
  edge_kernel<<<blocksE, kWaves * 32, 0, stream>>>(mji, co, eidx, Wa1, ba1, Wa2, ba2,
                                                   Wx1, bx1, Wx2, bx2,
                                                   a_out, m_acc, d_acc, E_, tilesE);

  const int tilesN  = N_ / kTM;
  const int blocksN = (tilesN + kWaves - 1) / kWaves;
  node_kernel<<<blocksN, kWaves * 32, 0, stream>>>(nf, co, Wn1, bn1, Wn2, bn2,
                                                   m_acc, d_acc, nf_out, co_out,
                                                   N_, tilesN);
}